// NTMWriteHead_1503238553613
// MI455X (gfx1250) — compile-verified
//
#include <hip/hip_runtime.h>
#include <math.h>

#define B_   1024
#define N_   1024
#define M_   128
#define C_   1024
#define OUTF 390            // 3*M + 6
#define EPS  1e-16f

typedef __attribute__((ext_vector_type(2))) float v2f;
typedef __attribute__((ext_vector_type(4))) float v4f;
typedef __attribute__((ext_vector_type(8))) float v8f;

// ---------------------------------------------------------------------------
// Kernel 1: o = embeddings @ fc_w + fc_b  using V_WMMA_F32_16X16X4_F32
// One wave per 16x16 output tile. K stepped by 4.
// A layout: lane l, vgpr v  <- A[l%16][2*(l/16)+v]
// B layout: lane l, vgpr v  <- B[2*(l/16)+v][l%16]
// D layout: lane l, vgpr v  <- D[v + 8*(l/16)][l%16]
// ---------------------------------------------------------------------------
__global__ __launch_bounds__(256) void ntm_fc_wmma(
    const float* __restrict__ E,      // (B, C) row-major
    const float* __restrict__ W,      // (C, OUTF) row-major
    const float* __restrict__ bias,   // (OUTF)
    float* __restrict__ O)            // (B, OUTF) row-major
{
    const int TILES_C = (OUTF + 15) / 16;     // 25
    const int TILES   = (B_ / 16) * TILES_C;  // 1600

    int wv   = threadIdx.x >> 5;
    int lane = threadIdx.x & 31;
    int tile = blockIdx.x * 8 + wv;
    if (tile >= TILES) return;

    int tr = tile / TILES_C;
    int tc = tile % TILES_C;
    int r0 = tr * 16;
    int c0 = tc * 16;

    int hf = lane >> 4;       // which half-wave
    int lm = lane & 15;

    int   col   = c0 + lm;
    int   colc  = (col < OUTF) ? col : (OUTF - 1);
    float cmask = (col < OUTF) ? 1.0f : 0.0f;

    const float* arow = E + (size_t)(r0 + lm) * C_ + 2 * hf;
    const float* bcol = W + (size_t)(2 * hf) * OUTF + colc;

    v8f acc = {};
    #pragma unroll 4
    for (int k0 = 0; k0 < C_; k0 += 4) {
        v2f a, b;
        a.x = arow[k0];
        a.y = arow[k0 + 1];
        b.x = bcol[(size_t)k0 * OUTF] * cmask;
        b.y = bcol[(size_t)(k0 + 1) * OUTF] * cmask;
        acc = __builtin_amdgcn_wmma_f32_16x16x4_f32(
                  false, a, false, b, (short)0, acc, false, false);
    }

    if (col < OUTF) {
        float bv = bias[col];
        #pragma unroll
        for (int v = 0; v < 8; ++v) {
            int row = r0 + v + 8 * hf;
            O[(size_t)row * OUTF + col] = acc[v] + bv;
        }
    }
}

// ---------------------------------------------------------------------------
// Wave32 / block reduction helpers
// ---------------------------------------------------------------------------
__device__ __forceinline__ float wave_sum32(float v) {
    #pragma unroll
    for (int off = 16; off; off >>= 1) v += __shfl_xor(v, off, 32);
    return v;
}

__device__ __forceinline__ float block_sum(float v, float* s_red, int tid) {
    #pragma unroll
    for (int off = 16; off; off >>= 1) v += __shfl_xor(v, off, 32);
    int wv = tid >> 5, lane = tid & 31;
    if (lane == 0) s_red[wv] = v;
    __syncthreads();
    if (tid < 32) {
        float x = (lane < 16) ? s_red[lane] : 0.0f;
        #pragma unroll
        for (int off = 8; off; off >>= 1) x += __shfl_xor(x, off, 32);
        if (lane == 0) s_red[0] = x;
    }
    __syncthreads();
    float r = s_red[0];
    __syncthreads();
    return r;
}

__device__ __forceinline__ float block_max(float v, float* s_red, int tid) {
    #pragma unroll
    for (int off = 16; off; off >>= 1) v = fmaxf(v, __shfl_xor(v, off, 32));
    int wv = tid >> 5, lane = tid & 31;
    if (lane == 0) s_red[wv] = v;
    __syncthreads();
    if (tid < 32) {
        float x = (lane < 16) ? s_red[lane] : -INFINITY;
        #pragma unroll
        for (int off = 8; off; off >>= 1) x = fmaxf(x, __shfl_xor(x, off, 32));
        if (lane == 0) s_red[0] = x;
    }
    __syncthreads();
    float r = s_red[0];
    __syncthreads();
    return r;
}

__device__ __forceinline__ float softplus_f(float x) {
    return fmaxf(x, 0.0f) + log1pf(expf(-fabsf(x)));
}
__device__ __forceinline__ float sigmoid_f(float x) {
    return 1.0f / (1.0f + expf(-x));
}

// ---------------------------------------------------------------------------
// Kernel 2: one block per batch element b. 512 threads = 16 waves.
//   Phase A: activations + k/e/a into LDS
//   Phase B: dot(memory[b,n,:], k) and ||memory[b,n,:]||^2 (wave per row, RT)
//   Phase C: softmax -> gate -> shift -> sharpen -> normalize -> write w
//   Phase D: new_memory = memory * (1 - w*e) + w*a  (NT load + NT stream store)
// ---------------------------------------------------------------------------
__global__ __launch_bounds__(512) void ntm_write_main(
    const float* __restrict__ O,        // (B, OUTF)
    const float* __restrict__ w_prev,   // (B, N)
    const float* __restrict__ mem,      // (B, N, M)
    float* __restrict__ out_w,          // (B, N)
    float* __restrict__ out_mem)        // (B, N, M)
{
    __shared__ float s_k[M_];
    __shared__ float s_e[M_];
    __shared__ float s_a[M_];
    __shared__ float s_dot[N_];
    __shared__ float s_nrm[N_];
    __shared__ float s_w[N_];
    __shared__ float s_red[16];
    __shared__ float sc[7];   // beta, g, s0, s1, s2, gamma, k_norm

    const int tid = threadIdx.x;
    const int b   = blockIdx.x;
    const float* ob = O + (size_t)b * OUTF;

    // ---- Phase A -----------------------------------------------------------
    if (tid < M_) {
        s_k[tid] = ob[tid];
        s_e[tid] = sigmoid_f(ob[M_ + 6 + tid]);
        s_a[tid] = ob[2 * M_ + 6 + tid];
    }
    __syncthreads();
    if (tid == 0) {
        float kn2 = 0.0f;
        for (int i = 0; i < M_; ++i) kn2 += s_k[i] * s_k[i];
        sc[6] = sqrtf(kn2);
        sc[0] = softplus_f(ob[M_ + 0]);            // beta
        sc[1] = sigmoid_f(ob[M_ + 1]);             // g
        float a0 = ob[M_ + 2], a1 = ob[M_ + 3], a2 = ob[M_ + 4];
        float mx = fmaxf(a0, fmaxf(a1, a2));
        float e0 = expf(a0 - mx), e1 = expf(a1 - mx), e2 = expf(a2 - mx);
        float es = e0 + e1 + e2;
        sc[2] = e0 / es; sc[3] = e1 / es; sc[4] = e2 / es;
        sc[5] = 1.0f + softplus_f(ob[M_ + 5]);     // gamma
    }
    __syncthreads();

    const int wv   = tid >> 5;
    const int lane = tid & 31;

    // ---- Phase B: dot + squared norm, wave per row (regular-temporal) ------
    const v4f  kv        = ((const v4f*)s_k)[lane];
    const v4f* mrow_base = (const v4f*)(mem + (size_t)b * N_ * M_);

    for (int n = wv * 64; n < wv * 64 + 64; ++n) {
        v4f mv = mrow_base[(size_t)n * (M_ / 4) + lane];
        float dp = mv.x * kv.x + mv.y * kv.y + mv.z * kv.z + mv.w * kv.w;
        float np = mv.x * mv.x + mv.y * mv.y + mv.z * mv.z + mv.w * mv.w;
        dp = wave_sum32(dp);
        np = wave_sum32(np);
        if (lane == 0) { s_dot[n] = dp; s_nrm[n] = np; }
    }
    __syncthreads();

    // ---- Phase C: addressing chain (each thread owns rows tid, tid+512) ----
    const float beta = sc[0], g = sc[1];
    const float sh0 = sc[2], sh1 = sc[3], sh2 = sc[4];
    const float gamma = sc[5], kn = sc[6];

    const int n0 = tid, n1 = tid + 512;
    float x0 = beta * (s_dot[n0] / (sqrtf(s_nrm[n0]) * kn + EPS));
    float x1 = beta * (s_dot[n1] / (sqrtf(s_nrm[n1]) * kn + EPS));

    float mx  = block_max(fmaxf(x0, x1), s_red, tid);
    float ex0 = expf(x0 - mx);
    float ex1 = expf(x1 - mx);
    float sm  = block_sum(ex0 + ex1, s_red, tid);
    float inv = 1.0f / sm;

    const float* wp = w_prev + (size_t)b * N_;
    float wg0 = g * (ex0 * inv) + (1.0f - g) * wp[n0];
    float wg1 = g * (ex1 * inv) + (1.0f - g) * wp[n1];
    s_w[n0] = wg0;
    s_w[n1] = wg1;
    __syncthreads();

    // circular shift: w_tilde[n] = s0*w_g[n-1] + s1*w_g[n] + s2*w_g[n+1]
    float wt0 = sh0 * s_w[(n0 + N_ - 1) & (N_ - 1)] + sh1 * wg0
              + sh2 * s_w[(n0 + 1) & (N_ - 1)];
    float wt1 = sh0 * s_w[(n1 + N_ - 1) & (N_ - 1)] + sh1 * wg1
              + sh2 * s_w[(n1 + 1) & (N_ - 1)];

    float wpw0 = powf(wt0 + EPS, gamma);
    float wpw1 = powf(wt1 + EPS, gamma);
    float psum = block_sum(wpw0 + wpw1, s_red, tid);   // barriers inside
    float winv = 1.0f / (psum + EPS);
    float w0 = wpw0 * winv;
    float w1 = wpw1 * winv;

    s_w[n0] = w0;                      // safe: block_sum sync'd after reads
    s_w[n1] = w1;
    out_w[(size_t)b * N_ + n0] = w0;
    out_w[(size_t)b * N_ + n1] = w1;
    __syncthreads();

    // ---- Phase D: memory update, wave per row, non-temporal streaming ------
    const v4f ev = ((const v4f*)s_e)[lane];
    const v4f av = ((const v4f*)s_a)[lane];
    v4f* orow = (v4f*)(out_mem + (size_t)b * N_ * M_);

    for (int n = wv * 64; n < wv * 64 + 64; ++n) {
        float wn = s_w[n];
        v4f mv = __builtin_nontemporal_load(&mrow_base[(size_t)n * (M_ / 4) + lane]);
        v4f r;
        r.x = mv.x * (1.0f - wn * ev.x) + wn * av.x;
        r.y = mv.y * (1.0f - wn * ev.y) + wn * av.y;
        r.z = mv.z * (1.0f - wn * ev.z) + wn * av.z;
        r.w = mv.w * (1.0f - wn * ev.w) + wn * av.w;
        __builtin_nontemporal_store(r, &orow[(size_t)n * (M_ / 4) + lane]);
    }
}

// ---------------------------------------------------------------------------
extern "C" void kernel_launch(void* const* d_in, const int* in_sizes, int n_in,
                              void* d_out, int out_size, void* d_ws, size_t ws_size,
                              hipStream_t stream) {
    const float* emb   = (const float*)d_in[0];   // (B, C)
    const float* wprev = (const float*)d_in[1];   // (B, N)
    const float* mem   = (const float*)d_in[2];   // (B, N, M)
    const float* fcw   = (const float*)d_in[3];   // (C, OUTF)
    const float* fcb   = (const float*)d_in[4];   // (OUTF)

    float* out     = (float*)d_out;
    float* out_w   = out;                          // first output: w (B*N)
    float* out_mem = out + (size_t)B_ * N_;        // second: new_memory

    float* o_ws = (float*)d_ws;                    // (B, OUTF) = 1.6 MB scratch

    const int TILES = (B_ / 16) * ((OUTF + 15) / 16);   // 1600 tiles
    ntm_fc_wmma<<<(TILES + 7) / 8, 256, 0, stream>>>(emb, fcw, fcb, o_ws);
    ntm_write_main<<<B_, 512, 0, stream>>>(o_ws, wprev, mem, out_w, out_mem);
}